// FourierRingCorrelation_37752762532604
// MI455X (gfx1250) — compile-verified
//
#include <hip/hip_runtime.h>

// ---------------------------------------------------------------------------
// Fourier Ring Correlation for MI455X (gfx1250, wave32).
//
// Stage 1 (bandwidth-bound, ~11.5us floor at 23.3 TB/s):
//   grid = 64 images x 16 blocks, 256 threads (8 waves / block).
//   Each thread streams 16B vectors (global_load_b128, non-temporal) from the
//   four input planes, computes ring bins on the fly (no index-array traffic),
//   accumulates 4 quantities into a 4KB LDS histogram via ds_add_f32 atomics
//   (merging the 4-pixels-same-ring common case), then writes its block's
//   1024-float partial histogram to d_ws (coalesced stores, no global atomics).
//
// Stage 2 (tiny, 4 MiB): per image, reduce 16 partial histograms:
//   acc[j] = sum_k part[k][j]  ==  ones(16x4-chained) x part(16x1024)
//   implemented with chained V_WMMA_F32_16X16X4_F32 (f32 accumulate, exact),
//   then finalize |cross| / sqrt(p1*p2) with den==0 -> 0.
// ---------------------------------------------------------------------------

#define B_    64
#define H_    512
#define W_    512
#define HW    (H_ * W_)
#define NRING 256
#define HBINS (4 * NRING)          // [cr, ci, p1, p2] x 256 rings = 1024

#define BLOCKS_PER_IMG 16
#define THREADS        256
#define PIX_PER_BLOCK  (HW / BLOCKS_PER_IMG)   // 16384
#define V4_PER_BLOCK   (PIX_PER_BLOCK / 4)     // 4096
#define ITERS          (V4_PER_BLOCK / THREADS) // 16

typedef __attribute__((ext_vector_type(2))) float v2f;
typedef __attribute__((ext_vector_type(4))) float v4f;
typedef __attribute__((ext_vector_type(8))) float v8f;

// ---------------------------------------------------------------------------
// Stage 1: streaming segmented reduction into per-block partial histograms.
// ---------------------------------------------------------------------------
__global__ __launch_bounds__(THREADS) void frc_partial(
    const float* __restrict__ f1r, const float* __restrict__ f1i,
    const float* __restrict__ f2r, const float* __restrict__ f2i,
    float* __restrict__ part)
{
    __shared__ float hist[HBINS];

    const int t   = threadIdx.x;
    const int img = blockIdx.x / BLOCKS_PER_IMG;
    const int blk = blockIdx.x % BLOCKS_PER_IMG;

    #pragma unroll
    for (int i = t; i < HBINS; i += THREADS) hist[i] = 0.0f;
    __syncthreads();

    const size_t base4 = (size_t)img * (HW / 4) + (size_t)blk * V4_PER_BLOCK;
    const v4f* a1r = (const v4f*)f1r + base4;
    const v4f* a1i = (const v4f*)f1i + base4;
    const v4f* a2r = (const v4f*)f2r + base4;
    const v4f* a2i = (const v4f*)f2i + base4;

    #pragma unroll 4
    for (int i = 0; i < ITERS; ++i) {
        const int idx4 = i * THREADS + t;
        // Non-temporal streaming loads: data is touched exactly once.
        const v4f x1r = __builtin_nontemporal_load(&a1r[idx4]);
        const v4f x1i = __builtin_nontemporal_load(&a1i[idx4]);
        const v4f x2r = __builtin_nontemporal_load(&a2r[idx4]);
        const v4f x2i = __builtin_nontemporal_load(&a2i[idx4]);

        const int pix = blk * PIX_PER_BLOCK + idx4 * 4;   // first of 4 pixels
        const int y   = pix >> 9;                          // W = 512
        const int x0  = pix & 511;                         // pix % 4 == 0
        const int dy  = y - 256;
        const int dy2 = dy * dy;
        const int dx  = x0 - 256;

        // Ring bins for 4 consecutive x (same row). floor(sqrtf) is exact
        // for r2 < 2^24 with this geometry (see analysis).
        const int b0 = (int)__builtin_sqrtf((float)(dx * dx + dy2));
        const int b1 = (int)__builtin_sqrtf((float)((dx + 1) * (dx + 1) + dy2));
        const int b2 = (int)__builtin_sqrtf((float)((dx + 2) * (dx + 2) + dy2));
        const int b3 = (int)__builtin_sqrtf((float)((dx + 3) * (dx + 3) + dy2));

        const float cr0 = x1r.x * x2r.x + x1i.x * x2i.x;
        const float cr1 = x1r.y * x2r.y + x1i.y * x2i.y;
        const float cr2 = x1r.z * x2r.z + x1i.z * x2i.z;
        const float cr3 = x1r.w * x2r.w + x1i.w * x2i.w;
        const float ci0 = x1i.x * x2r.x - x1r.x * x2i.x;
        const float ci1 = x1i.y * x2r.y - x1r.y * x2i.y;
        const float ci2 = x1i.z * x2r.z - x1r.z * x2i.z;
        const float ci3 = x1i.w * x2r.w - x1r.w * x2i.w;
        const float p10 = x1r.x * x1r.x + x1i.x * x1i.x;
        const float p11 = x1r.y * x1r.y + x1i.y * x1i.y;
        const float p12 = x1r.z * x1r.z + x1i.z * x1i.z;
        const float p13 = x1r.w * x1r.w + x1i.w * x1i.w;
        const float p20 = x2r.x * x2r.x + x2i.x * x2i.x;
        const float p21 = x2r.y * x2r.y + x2i.y * x2i.y;
        const float p22 = x2r.z * x2r.z + x2i.z * x2i.z;
        const float p23 = x2r.w * x2r.w + x2i.w * x2i.w;

        if ((b0 == b1) & (b1 == b2) & (b2 == b3)) {
            // Common case: all 4 pixels in the same ring -> 4 LDS atomics.
            if (b0 < NRING) {
                atomicAdd(&hist[          b0], cr0 + cr1 + cr2 + cr3);
                atomicAdd(&hist[NRING   + b0], ci0 + ci1 + ci2 + ci3);
                atomicAdd(&hist[2*NRING + b0], p10 + p11 + p12 + p13);
                atomicAdd(&hist[3*NRING + b0], p20 + p21 + p22 + p23);
            }
        } else {
            if (b0 < NRING) {
                atomicAdd(&hist[b0], cr0); atomicAdd(&hist[NRING + b0], ci0);
                atomicAdd(&hist[2*NRING + b0], p10); atomicAdd(&hist[3*NRING + b0], p20);
            }
            if (b1 < NRING) {
                atomicAdd(&hist[b1], cr1); atomicAdd(&hist[NRING + b1], ci1);
                atomicAdd(&hist[2*NRING + b1], p11); atomicAdd(&hist[3*NRING + b1], p21);
            }
            if (b2 < NRING) {
                atomicAdd(&hist[b2], cr2); atomicAdd(&hist[NRING + b2], ci2);
                atomicAdd(&hist[2*NRING + b2], p12); atomicAdd(&hist[3*NRING + b2], p22);
            }
            if (b3 < NRING) {
                atomicAdd(&hist[b3], cr3); atomicAdd(&hist[NRING + b3], ci3);
                atomicAdd(&hist[2*NRING + b3], p13); atomicAdd(&hist[3*NRING + b3], p23);
            }
        }
    }
    __syncthreads();

    float* dst = part + (size_t)(img * BLOCKS_PER_IMG + blk) * HBINS;
    #pragma unroll
    for (int i = t; i < HBINS; i += THREADS) dst[i] = hist[i];
}

// ---------------------------------------------------------------------------
// Stage 2: per image, acc[j] = sum_{k=0..15} part[k][j] via chained
// V_WMMA_F32_16X16X4_F32 (A = ones), then finalize FRC curve.
// One wave32 per image; EXEC is all-ones around the WMMAs.
// ---------------------------------------------------------------------------
__global__ __launch_bounds__(32) void frc_reduce(
    const float* __restrict__ part, float* __restrict__ out)
{
    __shared__ float acc[HBINS];

    const int lane = threadIdx.x;
    const int img  = blockIdx.x;
    const float* p = part + (size_t)img * BLOCKS_PER_IMG * HBINS;

    // A (16x4) all-ones: VGPR0 = {K0 | K2}, VGPR1 = {K1 | K3} -> just 1.0f.
    v2f aOnes;
    aOnes.x = 1.0f;
    aOnes.y = 1.0f;

    const int half = lane >> 4;    // 0: lanes 0-15 (K = kk+0,1), 1: lanes 16-31 (K = kk+2,3)
    const int nIdx = lane & 15;    // column within tile

    for (int tile = 0; tile < HBINS / 16; ++tile) {
        const int col = tile * 16 + nIdx;
        v8f c = {};
        #pragma unroll
        for (int kk = 0; kk < BLOCKS_PER_IMG; kk += 4) {
            // B (4x16): VGPR0 lanes0-15 = row K=kk+0, lanes16-31 = row K=kk+2
            //           VGPR1 lanes0-15 = row K=kk+1, lanes16-31 = row K=kk+3
            v2f b;
            b.x = p[(size_t)(kk + 0 + 2 * half) * HBINS + col];
            b.y = p[(size_t)(kk + 1 + 2 * half) * HBINS + col];
            c = __builtin_amdgcn_wmma_f32_16x16x4_f32(
                    /*neg_a=*/false, aOnes, /*neg_b=*/false, b,
                    /*c_mod=*/(short)0, c, /*reuse_a=*/false, /*reuse_b=*/false);
        }
        // D row M=0 lives in element 0, lanes 0-15 (N = lane).
        if (lane < 16) acc[col] = c[0];
    }
    __syncthreads();

    #pragma unroll
    for (int rr = 0; rr < NRING; rr += 32) {
        const int r = rr + lane;
        const float cr = acc[r];
        const float ci = acc[NRING + r];
        const float p1 = acc[2 * NRING + r];
        const float p2 = acc[3 * NRING + r];
        const float num = __builtin_sqrtf(cr * cr + ci * ci);
        const float den = __builtin_sqrtf(p1 * p2);
        out[(size_t)img * NRING + r] = (den == 0.0f) ? 0.0f : num / den;
    }
}

// ---------------------------------------------------------------------------
extern "C" void kernel_launch(void* const* d_in, const int* in_sizes, int n_in,
                              void* d_out, int out_size, void* d_ws, size_t ws_size,
                              hipStream_t stream)
{
    (void)in_sizes; (void)n_in; (void)out_size; (void)ws_size;
    const float* f1r = (const float*)d_in[0];
    const float* f1i = (const float*)d_in[1];
    const float* f2r = (const float*)d_in[2];
    const float* f2i = (const float*)d_in[3];
    float* part = (float*)d_ws;   // needs 64*16*1024*4 B = 4 MiB, fully
                                  // overwritten each call (deterministic).

    frc_partial<<<dim3(B_ * BLOCKS_PER_IMG), dim3(THREADS), 0, stream>>>(
        f1r, f1i, f2r, f2i, part);
    frc_reduce<<<dim3(B_), dim3(32), 0, stream>>>(part, (float*)d_out);
}